// EdgeGraphConv_489626272405
// MI455X (gfx1250) — compile-verified
//
#include <hip/hip_runtime.h>
#include <hip/hip_bf16.h>

typedef __attribute__((ext_vector_type(2))) float v2f;
typedef __attribute__((ext_vector_type(8))) float v8f;

#define IN_FEATS 128
#define OUT_FEATS 256
#define NTILES_PER_WAVE 4   // each wave computes a 16x64 output strip

// ---------------------------------------------------------------------------
// Kernel 1: zero the aggregation buffer (float4 stores, fully coalesced).
// ---------------------------------------------------------------------------
__global__ __launch_bounds__(256) void zero_f32_kernel(float* __restrict__ p, long n_floats) {
    long i = ((long)blockIdx.x * blockDim.x + threadIdx.x) * 4;
    if (i + 3 < n_floats) {
        *reinterpret_cast<float4*>(p + i) = make_float4(0.f, 0.f, 0.f, 0.f);
    } else {
        for (long j = i; j < n_floats; ++j) p[j] = 0.f;
    }
}

// ---------------------------------------------------------------------------
// Kernel 2: scatter-add edges into node accumulator.
// One wave (32 lanes) per edge; each lane owns a float4 of the 128-f32 row.
// feat row load is one coalesced B128 per lane; the 51.2 MB accumulator fits
// in the 192 MB L2, so the f32 atomics resolve on-chip.
// ---------------------------------------------------------------------------
__global__ __launch_bounds__(256) void scatter_add_kernel(
    const float* __restrict__ feat, const int* __restrict__ dst,
    float* __restrict__ agg, long n_edges) {
    long t = (long)blockIdx.x * blockDim.x + threadIdx.x;
    long edge = t >> 5;          // 32 lanes per edge
    int  lane = (int)(t & 31);
    if (edge >= n_edges) return;

    int node = dst[edge];
    const float4 v = *reinterpret_cast<const float4*>(
        feat + edge * (long)IN_FEATS + lane * 4);
    float* a = agg + (long)node * IN_FEATS + lane * 4;
    atomicAdd(a + 0, v.x);
    atomicAdd(a + 1, v.y);
    atomicAdd(a + 2, v.z);
    atomicAdd(a + 3, v.w);
}

// ---------------------------------------------------------------------------
// Kernel 3: out = relu(agg @ W^T + b) via V_WMMA_F32_16X16X4_F32 (exact fp32).
// One wave32 per 16x64 output strip (4 N-tiles): each A load (one B64/lane)
// feeds 4 WMMAs, cutting L2 reread of agg by 4x.
//
// CDNA5 ISA layouts (CDNA5 ISA):
//   A 16x4 f32:  lanes 0-15 row M=lane {v0=K0,v1=K1}; lanes 16-31 {v0=K2,v1=K3}.
//   B 4x16 f32:  lanes 0-15 col N=lane {v0=K0,v1=K1}; lanes 16-31 {v0=K2,v1=K3}.
//   C/D 16x16:   VGPR r, lanes 0-15 -> M=r, lanes 16-31 -> M=r+8, N=lane&15.
// B[k][n] = W[n][k] (W row-major [256,128]) => B columns are W rows.
// ---------------------------------------------------------------------------
__global__ __launch_bounds__(256) void gemm_bias_relu_kernel(
    const float* __restrict__ agg, const float* __restrict__ W,
    const float* __restrict__ bias, float* __restrict__ out, int n_nodes) {
    const int lane = threadIdx.x;                    // 0..31
    const int wave = threadIdx.y;                    // 0..7
    const int nGroups = OUT_FEATS / (16 * NTILES_PER_WAVE);  // 4

    int strip = blockIdx.x * 8 + wave;
    int mTile = strip / nGroups;
    int g     = strip - mTile * nGroups;
    int m0 = mTile * 16;
    int n0 = g * 16 * NTILES_PER_WAVE;
    if (m0 >= n_nodes) return;

    const int half = lane >> 4;          // 0: K0,K1   1: K2,K3
    const int l15  = lane & 15;

    // Clamp A row for the ragged M tail (EXEC must stay all-1s across WMMA).
    int arow = m0 + l15;
    if (arow >= n_nodes) arow = 0;

    const float* Ap = agg + (long)arow * IN_FEATS + half * 2;
    const float* Bp0 = W + (long)(n0 +  0 + l15) * IN_FEATS + half * 2;
    const float* Bp1 = W + (long)(n0 + 16 + l15) * IN_FEATS + half * 2;
    const float* Bp2 = W + (long)(n0 + 32 + l15) * IN_FEATS + half * 2;
    const float* Bp3 = W + (long)(n0 + 48 + l15) * IN_FEATS + half * 2;

    v8f c0 = {}, c1 = {}, c2 = {}, c3 = {};
#pragma unroll 2
    for (int k = 0; k < IN_FEATS; k += 4) {
        v2f a  = *reinterpret_cast<const v2f*>(Ap  + k);
        v2f b0 = *reinterpret_cast<const v2f*>(Bp0 + k);
        v2f b1 = *reinterpret_cast<const v2f*>(Bp1 + k);
        v2f b2 = *reinterpret_cast<const v2f*>(Bp2 + k);
        v2f b3 = *reinterpret_cast<const v2f*>(Bp3 + k);
        // 8 args: (neg_a, A, neg_b, B, c_mod, C, reuse_a, reuse_b)
        c0 = __builtin_amdgcn_wmma_f32_16x16x4_f32(false, a, false, b0, (short)0, c0, false, false);
        c1 = __builtin_amdgcn_wmma_f32_16x16x4_f32(false, a, false, b1, (short)0, c1, false, false);
        c2 = __builtin_amdgcn_wmma_f32_16x16x4_f32(false, a, false, b2, (short)0, c2, false, false);
        c3 = __builtin_amdgcn_wmma_f32_16x16x4_f32(false, a, false, b3, (short)0, c3, false, false);
    }

    // Epilogue: bias + relu + store.
    // VGPR r of tile j holds row M = m0 + r + half*8, column N = n0 + j*16 + l15.
    const bool fullTile = (m0 + 16 <= n_nodes);      // wave-uniform branch
    float* outRow = out + (long)(m0 + half * 8) * OUT_FEATS + l15;

    v8f cs[NTILES_PER_WAVE] = {c0, c1, c2, c3};
#pragma unroll
    for (int j = 0; j < NTILES_PER_WAVE; ++j) {
        int ncol = n0 + j * 16;
        float bv = bias[ncol + l15];
        float* outp = outRow + ncol;
        if (fullTile) {
#pragma unroll
            for (int r = 0; r < 8; ++r) {
                float v = cs[j][r] + bv;
                outp[(long)r * OUT_FEATS] = v > 0.f ? v : 0.f;
            }
        } else {
#pragma unroll
            for (int r = 0; r < 8; ++r) {
                if (m0 + half * 8 + r < n_nodes) {
                    float v = cs[j][r] + bv;
                    outp[(long)r * OUT_FEATS] = v > 0.f ? v : 0.f;
                }
            }
        }
    }
}

// ---------------------------------------------------------------------------
// Launch
// ---------------------------------------------------------------------------
extern "C" void kernel_launch(void* const* d_in, const int* in_sizes, int n_in,
                              void* d_out, int out_size, void* d_ws, size_t ws_size,
                              hipStream_t stream) {
    const float* feat = (const float*)d_in[0];
    const int*   dst  = (const int*)d_in[1];
    const float* W    = (const float*)d_in[2];
    const float* b    = (const float*)d_in[3];
    float*       out  = (float*)d_out;
    float*       agg  = (float*)d_ws;   // n_nodes * 128 floats

    const long n_edges = (long)in_sizes[0] / IN_FEATS;
    const int  n_nodes = out_size / OUT_FEATS;   // derived on host (no device reads in capture)

    // 1) zero accumulator (every call, for determinism)
    {
        long n_floats = (long)n_nodes * IN_FEATS;
        long threads  = (n_floats + 3) / 4;
        int  blocks   = (int)((threads + 255) / 256);
        zero_f32_kernel<<<blocks, 256, 0, stream>>>(agg, n_floats);
    }

    // 2) scatter-add edges -> nodes
    {
        long threads = n_edges * 32;
        int  blocks  = (int)((threads + 255) / 256);
        scatter_add_kernel<<<blocks, 256, 0, stream>>>(feat, dst, agg, n_edges);
    }

    // 3) WMMA GEMM + bias + relu (each wave: 16x64 strip)
    {
        int mTiles  = (n_nodes + 15) / 16;
        int nGroups = OUT_FEATS / (16 * NTILES_PER_WAVE); // 4
        int waves   = mTiles * nGroups;
        int blocks  = (waves + 7) / 8;   // 8 waves (256 threads) per block
        dim3 blk(32, 8, 1);
        gemm_bias_relu_kernel<<<blocks, blk, 0, stream>>>(agg, W, b, out, n_nodes);
    }
}